// MNLoss_25769803776122
// MI455X (gfx1250) — compile-verified
//
#include <hip/hip_runtime.h>

#define BLOCK_THREADS 256
#define WAVE_SIZE 32
#define WAVES_PER_BLOCK (BLOCK_THREADS / WAVE_SIZE)
#define MAX_KF 128  // max floats/row handled by the LDS staging path (K=100 here)

#define ALIGNED_MARGIN 0.001f
#define MISALIGNED_SHIFT 0.15f
#define LEAKY_SLOPE 0.01f

// ---- CDNA5 async global->LDS path (probe via __has_builtin) -----------------
#if __has_builtin(__builtin_amdgcn_global_load_async_to_lds_b128)
#define HAVE_ASYNC_LDS 1
// Diagnostic from round 1 shows param 0 is: v4i (vector_size(16) int) pointer
// in addrspace(1) (clang prints AS1 as `__device__`). Match it exactly.
typedef int v4i __attribute__((vector_size(16)));
typedef __attribute__((address_space(1))) v4i* gptr_v4i;
typedef __attribute__((address_space(3))) v4i* lptr_v4i;

__device__ __forceinline__ void async_copy16(const float* g, float* l) {
  // global_load_async_to_lds_b128: per-lane 16B global -> LDS, tracked by ASYNCcnt
  __builtin_amdgcn_global_load_async_to_lds_b128((gptr_v4i)g, (lptr_v4i)l, 0, 0);
}
#endif

__device__ __forceinline__ void wait_async_le1() {
#if __has_builtin(__builtin_amdgcn_s_wait_asynccnt)
  __builtin_amdgcn_s_wait_asynccnt(1);
#else
  asm volatile("s_wait_asynccnt 0x1" ::: "memory");
#endif
}
__device__ __forceinline__ void wait_async_le0() {
#if __has_builtin(__builtin_amdgcn_s_wait_asynccnt)
  __builtin_amdgcn_s_wait_asynccnt(0);
#else
  asm volatile("s_wait_asynccnt 0x0" ::: "memory");
#endif
}

// ---- per-row math, shared by both data paths --------------------------------
__device__ __forceinline__ void accum_chunk(float4 v, int base, int len,
                                            float& s, float& m) {
  if (base + 0 < len) { s += fmaxf(ALIGNED_MARGIN - v.x, 0.0f); m = fminf(m, v.x); }
  if (base + 1 < len) { s += fmaxf(ALIGNED_MARGIN - v.y, 0.0f); m = fminf(m, v.y); }
  if (base + 2 < len) { s += fmaxf(ALIGNED_MARGIN - v.z, 0.0f); m = fminf(m, v.z); }
  if (base + 3 < len) { s += fmaxf(ALIGNED_MARGIN - v.w, 0.0f); m = fminf(m, v.w); }
}

__device__ __forceinline__ float finish_row(float s, float m, int lab, int len) {
  // wave32 xor-tree reduction: sum (aligned branch) + min (misaligned branch)
#pragma unroll
  for (int off = 16; off > 0; off >>= 1) {
    s += __shfl_xor(s, off, WAVE_SIZE);
    m = fminf(m, __shfl_xor(m, off, WAVE_SIZE));
  }
  if (lab == 1) {
    return s / (float)len;
  } else {
    float t = m + MISALIGNED_SHIFT;
    return t > 0.0f ? t : LEAKY_SLOPE * t;
  }
}

// ---- stage 1: per-block partial sums ----------------------------------------
__global__ __launch_bounds__(BLOCK_THREADS) void mnloss_partial_kernel(
    const float* __restrict__ sim, const int* __restrict__ label,
    const int* __restrict__ lens, float* __restrict__ partial, int N, int K) {
#ifdef HAVE_ASYNC_LDS
  __shared__ __align__(16) float stage[WAVES_PER_BLOCK][2][MAX_KF];
#endif
  __shared__ float wsum[WAVES_PER_BLOCK];

  const int lane = threadIdx.x & (WAVE_SIZE - 1);
  const int wid = threadIdx.x / WAVE_SIZE;
  const int gw = blockIdx.x * WAVES_PER_BLOCK + wid;
  const int wstride = gridDim.x * WAVES_PER_BLOCK;
  const int KV = K >> 2;  // float4 chunks per row (K is a multiple of 4)

  float acc = 0.0f;

#ifdef HAVE_ASYNC_LDS
  if (K <= MAX_KF && (K & 3) == 0) {
    // double-buffered async global->LDS streaming pipeline (ASYNCcnt)
    int r = gw;
    int buf = 0;
    if (r < N && lane < KV)
      async_copy16(sim + (size_t)r * K + lane * 4, &stage[wid][0][lane * 4]);
    while (r < N) {
      const int rn = r + wstride;
      if (rn < N) {
        if (lane < KV)
          async_copy16(sim + (size_t)rn * K + lane * 4,
                       &stage[wid][buf ^ 1][lane * 4]);
        wait_async_le1();  // row r landed in LDS; next row in flight
      } else {
        wait_async_le0();
      }
      const int len = lens[r];
      const int lab = label[r];
      float s = 0.0f, m = __builtin_inff();
      if (lane < KV) {
        const float4 v =
            *reinterpret_cast<const float4*>(&stage[wid][buf][lane * 4]);
        accum_chunk(v, lane * 4, len, s, m);
      }
      acc += finish_row(s, m, lab, len);
      buf ^= 1;
      r = rn;
    }
  } else
#endif
  {
    // direct coalesced float4 loads + explicit prefetch of the next row
    for (int r = gw; r < N; r += wstride) {
      const int len = lens[r];
      const int lab = label[r];
      const float4* row = reinterpret_cast<const float4*>(sim + (size_t)r * K);
      if (r + wstride < N)
        __builtin_prefetch(sim + (size_t)(r + wstride) * K + lane * 4, 0, 0);
      float s = 0.0f, m = __builtin_inff();
      for (int c = lane; c < KV; c += WAVE_SIZE) {
        accum_chunk(row[c], c * 4, len, s, m);
      }
      acc += finish_row(s, m, lab, len);
    }
  }

  if (lane == 0) wsum[wid] = acc;
  __syncthreads();
  if (threadIdx.x == 0) {
    float b = 0.0f;
#pragma unroll
    for (int i = 0; i < WAVES_PER_BLOCK; ++i) b += wsum[i];
    partial[blockIdx.x] = b;
  }
}

// ---- stage 2: reduce block partials deterministically -----------------------
__global__ __launch_bounds__(BLOCK_THREADS) void mnloss_reduce_kernel(
    const float* __restrict__ partial, int n, float* __restrict__ out) {
  __shared__ float sm[BLOCK_THREADS];
  float a = 0.0f;
  for (int i = threadIdx.x; i < n; i += BLOCK_THREADS) a += partial[i];
  sm[threadIdx.x] = a;
  __syncthreads();
  for (int s = BLOCK_THREADS / 2; s > 0; s >>= 1) {
    if ((int)threadIdx.x < s) sm[threadIdx.x] += sm[threadIdx.x + s];
    __syncthreads();
  }
  if (threadIdx.x == 0) out[0] = sm[0];
}

extern "C" void kernel_launch(void* const* d_in, const int* in_sizes, int n_in,
                              void* d_out, int out_size, void* d_ws,
                              size_t ws_size, hipStream_t stream) {
  const float* sim = (const float*)d_in[0];
  const int* label = (const int*)d_in[1];
  const int* lens = (const int*)d_in[2];
  float* out = (float*)d_out;

  const int N = in_sizes[1];                // 1,000,000
  const int K = in_sizes[0] / in_sizes[1];  // 100

  int nb = 1024;
  const size_t max_partials = ws_size / sizeof(float);
  if ((size_t)nb > max_partials) nb = (int)max_partials;
  if (nb < 1) nb = 1;
  float* partial = (float*)d_ws;

  mnloss_partial_kernel<<<nb, BLOCK_THREADS, 0, stream>>>(sim, label, lens,
                                                          partial, N, K);
  mnloss_reduce_kernel<<<1, BLOCK_THREADS, 0, stream>>>(partial, nb, out);
}